// GAT_5282809774861
// MI455X (gfx1250) — compile-verified
//
#include <hip/hip_runtime.h>
#include <stdint.h>

#define IN_CH 34
#define NEG_SLOPE 0.2f

typedef float v2f __attribute__((ext_vector_type(2)));
typedef float v8f __attribute__((ext_vector_type(8)));

// ---- helpers -------------------------------------------------------------
__device__ __forceinline__ int enc_f32(float f) {
    int i = __float_as_int(f);
    return i >= 0 ? i : (i ^ 0x7FFFFFFF);   // order-preserving int encoding
}
__device__ __forceinline__ float dec_f32(int i) {
    return __int_as_float(i >= 0 ? i : (i ^ 0x7FFFFFFF));
}
__device__ __forceinline__ float lrelu(float x) { return x > 0.f ? x : NEG_SLOPE * x; }

// ---- f32 WMMA GEMM: C[nrows,NCOLS] = A[nrows,KACT] @ B[KACT,NCOLS] -------
// One wave computes a 16 x (16*NTW) output strip with V_WMMA_F32_16X16X4_F32.
// Branch-free inner loop: unconditional b64 A loads, clamped+masked K remainder.
template<int KACT, int NCOLS, int NTW>
__global__ __launch_bounds__(128) void gat_wmma_gemm(
    const float* __restrict__ A, const float* __restrict__ B,
    float* __restrict__ C, int nrows, int ntiles) {
    constexpr int NGROUPS = NCOLS / (16 * NTW);
    const int lane = threadIdx.x & 31;
    const int wave = threadIdx.x >> 5;
    const int tile = blockIdx.x * 4 + wave;
    if (tile >= ntiles) return;                 // wave-uniform: EXEC stays all-1s
    const int mt = tile / NGROUPS;
    const int ng = tile - mt * NGROUPS;
    const int mbase  = mt * 16;
    const int nbase0 = ng * (16 * NTW);
    const int half = lane >> 4;                 // 0: lanes 0-15, 1: lanes 16-31
    const int l    = lane & 15;
    int row = mbase + l;
    row = row < nrows ? row : nrows - 1;        // clamp reads; stores guarded below
    const float* __restrict__ Arow = A + (size_t)row * KACT;

    v8f acc[NTW] = {};

    constexpr int KMAIN = KACT & ~3;
    #pragma unroll
    for (int k0 = 0; k0 < KMAIN; k0 += 4) {
        // A 16x4 f32: lanes 0-15 -> K=k0,k0+1 ; lanes 16-31 -> K=k0+2,k0+3
        const int ka = k0 + half * 2;           // even => 8B-aligned (KACT even)
        const v2f a = *(const v2f*)(Arow + ka);
        #pragma unroll
        for (int j = 0; j < NTW; ++j) {
            const int col = nbase0 + 16 * j + l;
            v2f b;
            b.x = B[(size_t)ka * NCOLS + col];
            b.y = B[(size_t)(ka + 1) * NCOLS + col];
            acc[j] = __builtin_amdgcn_wmma_f32_16x16x4_f32(
                false, a, false, b, (short)0, acc[j], false, false);
        }
    }
    if constexpr ((KACT & 3) != 0) {            // K remainder, branch-free
        const int ka = KMAIN + half * 2;
        const int kb = ka + 1;
        const float ma = (ka < KACT) ? 1.f : 0.f;
        const float mb = (kb < KACT) ? 1.f : 0.f;
        const int kac = (ka < KACT) ? ka : 0;   // clamped in-bounds index
        const int kbc = (kb < KACT) ? kb : 0;
        v2f a;
        a.x = Arow[kac] * ma;
        a.y = Arow[kbc] * mb;
        #pragma unroll
        for (int j = 0; j < NTW; ++j) {
            const int col = nbase0 + 16 * j + l;
            v2f b;
            b.x = B[(size_t)kac * NCOLS + col] * ma;
            b.y = B[(size_t)kbc * NCOLS + col] * mb;
            acc[j] = __builtin_amdgcn_wmma_f32_16x16x4_f32(
                false, a, false, b, (short)0, acc[j], false, false);
        }
    }
    // C/D layout: VGPR r -> M=r (lanes 0-15), M=r+8 (lanes 16-31); N = lane&15
    #pragma unroll
    for (int j = 0; j < NTW; ++j) {
        const int col = nbase0 + 16 * j + l;
        #pragma unroll
        for (int r = 0; r < 8; ++r) {
            const int orow = mbase + half * 8 + r;
            if (orow < nrows) C[(size_t)orow * NCOLS + col] = acc[j][r];
        }
    }
}

// ---- per-(node,head) attention logits -----------------------------------
template<int H>
__global__ void gat_compute_al(const float* __restrict__ xw,
                               const float* __restrict__ a_src,
                               const float* __restrict__ a_dst,
                               float* __restrict__ al_s,
                               float* __restrict__ al_d, int n) {
    const int idx = blockIdx.x * blockDim.x + threadIdx.x;
    if (idx >= n * H) return;
    const int h = idx % H;
    const float* __restrict__ row = xw + (size_t)(idx / H) * (H * 64) + h * 64;
    const float* __restrict__ as  = a_src + h * 64;
    const float* __restrict__ ad  = a_dst + h * 64;
    float ss = 0.f, sd = 0.f;
    #pragma unroll 8
    for (int c = 0; c < 64; ++c) {
        const float v = row[c];
        ss = fmaf(v, as[c], ss);
        sd = fmaf(v, ad[c], sd);
    }
    al_s[idx] = ss;
    al_d[idx] = sd;
}

// ---- edge pass 1: segment max (sortable-int atomicMax) -------------------
template<int H>
__global__ void gat_edge_max(const long long* __restrict__ esrc,
                             const long long* __restrict__ edst,
                             const float* __restrict__ al_s,
                             const float* __restrict__ al_d,
                             int* __restrict__ menc, int E_real, int n_tot) {
    const int eid = blockIdx.x * blockDim.x + threadIdx.x;
    if (eid >= n_tot) return;
    int s, d;
    if (eid < E_real) { s = (int)esrc[eid]; d = (int)edst[eid]; }
    else              { s = d = eid - E_real; }            // self loop
    #pragma unroll
    for (int h = 0; h < H; ++h) {
        const float e = lrelu(al_s[s * H + h] + al_d[d * H + h]);
        atomicMax(&menc[d * H + h], enc_f32(e));
    }
}

// ---- edge pass 2: segment sum of exp(e - m) ------------------------------
template<int H>
__global__ void gat_edge_sum(const long long* __restrict__ esrc,
                             const long long* __restrict__ edst,
                             const float* __restrict__ al_s,
                             const float* __restrict__ al_d,
                             const int* __restrict__ menc,
                             float* __restrict__ den, int E_real, int n_tot) {
    const int eid = blockIdx.x * blockDim.x + threadIdx.x;
    if (eid >= n_tot) return;
    int s, d;
    if (eid < E_real) { s = (int)esrc[eid]; d = (int)edst[eid]; }
    else              { s = d = eid - E_real; }
    #pragma unroll
    for (int h = 0; h < H; ++h) {
        const float e = lrelu(al_s[s * H + h] + al_d[d * H + h]);
        const float m = dec_f32(menc[d * H + h]);
        atomicAdd(&den[d * H + h], __expf(e - m));
    }
}

// ---- edge pass 3: out[dst] += alpha * xw[src], wave per edge -------------
template<int H, int C>
__global__ void gat_edge_msg(const long long* __restrict__ esrc,
                             const long long* __restrict__ edst,
                             const float* __restrict__ al_s,
                             const float* __restrict__ al_d,
                             const int* __restrict__ menc,
                             const float* __restrict__ den,
                             const float* __restrict__ xw,
                             float* __restrict__ out, int E_real, int n_tot) {
    const int lane = threadIdx.x & 31;
    const int wid  = blockIdx.x * (blockDim.x >> 5) + (threadIdx.x >> 5);
    if (wid >= n_tot) return;
    int s, d;
    if (wid < E_real) { s = (int)esrc[wid]; d = (int)edst[wid]; }
    else              { s = d = wid - E_real; }
    #pragma unroll
    for (int h = 0; h < H; ++h) {
        const float e     = lrelu(al_s[s * H + h] + al_d[d * H + h]);
        const float alpha = __expf(e - dec_f32(menc[d * H + h])) / den[d * H + h];
        const float* __restrict__ xs = xw  + (size_t)s * (H * C) + h * C;
        float*       __restrict__ od = out + (size_t)d * (H * C) + h * C;
        #pragma unroll
        for (int c0 = lane * 2; c0 < C; c0 += 64) {    // 8B-aligned pair per lane
            const v2f v = *(const v2f*)(xs + c0);
            atomicAdd(&od[c0],     v.x * alpha);
            atomicAdd(&od[c0 + 1], v.y * alpha);
        }
    }
}

// ---- utility kernels -----------------------------------------------------
__global__ void gat_fill_u32x4(uint4* __restrict__ p, unsigned v, int n4) {
    const int i = blockIdx.x * blockDim.x + threadIdx.x;
    if (i < n4) p[i] = make_uint4(v, v, v, v);
}
__global__ void gat_bias_act(float* __restrict__ p, const float* __restrict__ bias,
                             int n, int stride, int do_relu) {
    const int i = blockIdx.x * blockDim.x + threadIdx.x;
    if (i >= n) return;
    float v = p[i] + bias[i & (stride - 1)];    // stride is a power of two
    if (do_relu) v = fmaxf(v, 0.f);
    p[i] = v;
}

// ---- launch --------------------------------------------------------------
extern "C" void kernel_launch(void* const* d_in, const int* in_sizes, int n_in,
                              void* d_out, int out_size, void* d_ws, size_t ws_size,
                              hipStream_t stream) {
    (void)n_in; (void)out_size; (void)ws_size;
    const float*     x    = (const float*)d_in[0];
    const long long* ei   = (const long long*)d_in[1];   // int64 [2,E]
    const float*     W1   = (const float*)d_in[2];
    const float*     as1  = (const float*)d_in[3];
    const float*     ad1  = (const float*)d_in[4];
    const float*     b1   = (const float*)d_in[5];
    const float*     W2   = (const float*)d_in[6];
    const float*     as2  = (const float*)d_in[7];
    const float*     ad2  = (const float*)d_in[8];
    const float*     b2   = (const float*)d_in[9];
    float*           out  = (float*)d_out;

    const int N  = in_sizes[0] / IN_CH;
    const int E  = in_sizes[1] / 2;
    const int ET = E + N;                       // edges + self loops
    const long long* esrc = ei;
    const long long* edst = ei + E;

    float* ws  = (float*)d_ws;
    float* xw1  = ws;  ws += (size_t)N * 128;   // x @ W1   [N,2,64]
    float* h1   = ws;  ws += (size_t)N * 128;   // layer-1 accum / relu'd hidden
    float* xw2  = ws;  ws += (size_t)N * 64;    // h @ W2   [N,1,64]
    float* al1s = ws;  ws += (size_t)N * 2;
    float* al1d = ws;  ws += (size_t)N * 2;
    float* al2s = ws;  ws += (size_t)N;
    float* al2d = ws;  ws += (size_t)N;
    int*   menc = (int*)ws; ws += (size_t)N * 2;
    float* den  = ws;  ws += (size_t)N * 2;

    const int TB = 256;
    auto cdiv = [](int a, int b) { return (a + b - 1) / b; };

    // ---------------- layer 1 (heads=2, concat) ----------------
    {
        const int ntiles = cdiv(N, 16) * (128 / 64);     // 16x64 strip per wave
        gat_wmma_gemm<IN_CH, 128, 4><<<cdiv(ntiles, 4), 128, 0, stream>>>(
            x, W1, xw1, N, ntiles);
    }
    gat_compute_al<2><<<cdiv(N * 2, TB), TB, 0, stream>>>(xw1, as1, ad1, al1s, al1d, N);
    gat_fill_u32x4<<<cdiv(N * 2 / 4, TB), TB, 0, stream>>>((uint4*)menc, 0x80000000u, N * 2 / 4);
    gat_fill_u32x4<<<cdiv(N * 2 / 4, TB), TB, 0, stream>>>((uint4*)den, 0u, N * 2 / 4);
    gat_fill_u32x4<<<cdiv(N * 32, TB), TB, 0, stream>>>((uint4*)h1, 0u, N * 32);
    gat_edge_max<2><<<cdiv(ET, TB), TB, 0, stream>>>(esrc, edst, al1s, al1d, menc, E, ET);
    gat_edge_sum<2><<<cdiv(ET, TB), TB, 0, stream>>>(esrc, edst, al1s, al1d, menc, den, E, ET);
    gat_edge_msg<2, 64><<<cdiv(ET, 8), TB, 0, stream>>>(esrc, edst, al1s, al1d, menc, den,
                                                        xw1, h1, E, ET);
    gat_bias_act<<<cdiv(N * 128, TB), TB, 0, stream>>>(h1, b1, N * 128, 128, 1);

    // ---------------- layer 2 (heads=1, mean == identity) ----------------
    {
        const int ntiles = cdiv(N, 16) * (64 / 64);
        gat_wmma_gemm<128, 64, 4><<<cdiv(ntiles, 4), 128, 0, stream>>>(
            h1, W2, xw2, N, ntiles);
    }
    gat_compute_al<1><<<cdiv(N, TB), TB, 0, stream>>>(xw2, as2, ad2, al2s, al2d, N);
    gat_fill_u32x4<<<cdiv(N / 4, TB), TB, 0, stream>>>((uint4*)menc, 0x80000000u, N / 4);
    gat_fill_u32x4<<<cdiv(N / 4, TB), TB, 0, stream>>>((uint4*)den, 0u, N / 4);
    gat_fill_u32x4<<<cdiv(N * 16, TB), TB, 0, stream>>>((uint4*)out, 0u, N * 16);
    gat_edge_max<1><<<cdiv(ET, TB), TB, 0, stream>>>(esrc, edst, al2s, al2d, menc, E, ET);
    gat_edge_sum<1><<<cdiv(ET, TB), TB, 0, stream>>>(esrc, edst, al2s, al2d, menc, den, E, ET);
    gat_edge_msg<1, 64><<<cdiv(ET, 8), TB, 0, stream>>>(esrc, edst, al2s, al2d, menc, den,
                                                        xw2, out, E, ET);
    gat_bias_act<<<cdiv(N * 64, TB), TB, 0, stream>>>(out, b2, N * 64, 64, 0);
}